// Model_18751827214746
// MI455X (gfx1250) — compile-verified
//
#include <hip/hip_runtime.h>
#include <math.h>

typedef float v2f __attribute__((ext_vector_type(2)));
typedef float v8f __attribute__((ext_vector_type(8)));

#define Tn 9
#define Hn 128
#define Wn 128
#define Bn 8
#define HW (Hn * Wn)            // 16384
#define PLANE (Tn * HW)         // 147456 per (sample,channel)
#define VOXSZ (Bn * 2 * PLANE)  // 2359296
#define H16SZ (Bn * 16 * PLANE) // 18874368

// workspace layout (float offsets); total ~104 MB
#define OFF_TMAX   0
#define OFF_VMEAN  8
#define OFF_VRSTD  24
#define OFF_POOL   40        // [B*16]
#define OFF_OMEAN  168
#define OFF_ORSTD  184
#define OFF_WDYN   256       // [B*16*16*3] = 6144
#define OFF_VOX    8192
#define OFF_AROUND (OFF_VOX + VOXSZ)
#define OFF_MASK   (OFF_AROUND + VOXSZ)
#define OFF_H16    (OFF_MASK + VOXSZ)

__global__ void k_zero(float* p, int n) {
  int i = blockIdx.x * blockDim.x + threadIdx.x;
  int st = gridDim.x * blockDim.x;
  for (; i < n; i += st) p[i] = 0.f;
}

// segment max of t per batch (t >= 0 so int-compare == float-compare)
__global__ void k_tmax(const float* t, const int* b, float* tmax, int n) {
  int i = blockIdx.x * blockDim.x + threadIdx.x;
  if (i < n) atomicMax((int*)(tmax + b[i]), __float_as_int(t[i]));
}

// bilinear-in-time scatter-add: each event touches at most 2 of the T bins
__global__ void k_vox(const float* t, const int* xi, const int* yi, const int* pi,
                      const int* bi, const float* tmax, float* vox, int n) {
  int i = blockIdx.x * blockDim.x + threadIdx.x;
  if (i >= n) return;
  int b = bi[i];
  float tm = tmax[b];
  float ts = (tm > 0.f) ? (t[i] / tm) * 9.f : 0.f;
  int base = xi[i] + Wn * yi[i] + PLANE * pi[i] + 2 * PLANE * b;
  int j0 = (int)floorf(ts);
#pragma unroll
  for (int jj = 0; jj < 2; ++jj) {
    int j = j0 + jj;
    if (j >= 0 && j < Tn) {
      float val = 1.f - fabsf(ts - (float)j);
      if (val > 0.f) atomicAdd(vox + base + j * HW, val);
    }
  }
}

// per-(b,c) mean + rsqrt(var+eps) over one contiguous plane
__global__ void k_stats(const float* src, float* meanOut, float* rstdOut, int plane) {
  const float* p = src + (size_t)blockIdx.x * plane;
  float s = 0.f, s2 = 0.f;
  for (int i = threadIdx.x; i < plane; i += blockDim.x) {
    float v = p[i];
    s += v; s2 += v * v;
  }
  __shared__ float sh[16];
  for (int o = 16; o; o >>= 1) { s += __shfl_xor(s, o, 32); s2 += __shfl_xor(s2, o, 32); }
  int lane = threadIdx.x & 31, wv = threadIdx.x >> 5;
  if (lane == 0) { sh[wv] = s; sh[8 + wv] = s2; }
  __syncthreads();
  if (threadIdx.x == 0) {
    float ts = 0.f, ts2 = 0.f;
    for (int w = 0; w < (int)(blockDim.x >> 5); ++w) { ts += sh[w]; ts2 += sh[8 + w]; }
    float m = ts / (float)plane;
    float var = ts2 / (float)plane - m * m;
    meanOut[blockIdx.x] = m;
    rstdOut[blockIdx.x] = rsqrtf(var + 1e-5f);
  }
}

__global__ void k_mean(const float* src, float* meanOut, int plane) {
  const float* p = src + (size_t)blockIdx.x * plane;
  float s = 0.f;
  for (int i = threadIdx.x; i < plane; i += blockDim.x) s += p[i];
  __shared__ float sh[8];
  for (int o = 16; o; o >>= 1) s += __shfl_xor(s, o, 32);
  int lane = threadIdx.x & 31, wv = threadIdx.x >> 5;
  if (lane == 0) sh[wv] = s;
  __syncthreads();
  if (threadIdx.x == 0) {
    float ts = 0.f;
    for (int w = 0; w < (int)(blockDim.x >> 5); ++w) ts += sh[w];
    meanOut[blockIdx.x] = ts / (float)plane;
  }
}

// around = bx[t+1] && (3x3 count of bx[t] >= 5)   (exact integer form of >4/9)
__global__ void k_around(const float* vox, float* around) {
  int idx = blockIdx.x * blockDim.x + threadIdx.x;
  if (idx >= VOXSZ) return;
  int x = idx & (Wn - 1);
  int y = (idx >> 7) & (Hn - 1);
  int tt = (idx / HW) % Tn;
  int bc = idx / PLANE;
  const float* sl  = vox + (size_t)bc * PLANE + (size_t)tt * HW;
  const float* sln = vox + (size_t)bc * PLANE + (size_t)((tt + 1 < Tn) ? tt + 1 : tt) * HW;
  int cnt = 0;
#pragma unroll
  for (int dy = -1; dy <= 1; ++dy)
#pragma unroll
    for (int dx = -1; dx <= 1; ++dx) {
      int yy = y + dy, xx = x + dx;
      if (yy >= 0 && yy < Hn && xx >= 0 && xx < Wn) cnt += (sl[yy * Wn + xx] > 0.f) ? 1 : 0;
    }
  int bn = (sln[y * Wn + x] > 0.f) ? 1 : 0;
  around[idx] = (bn && cnt >= 5) ? 1.f : 0.f;
}

// mask = 3x3 count_include_pad average of `around`
__global__ void k_mask(const float* around, float* mask) {
  int idx = blockIdx.x * blockDim.x + threadIdx.x;
  if (idx >= VOXSZ) return;
  int x = idx & (Wn - 1);
  int y = (idx >> 7) & (Hn - 1);
  int tt = (idx / HW) % Tn;
  int bc = idx / PLANE;
  const float* sl = around + (size_t)bc * PLANE + (size_t)tt * HW;
  float s = 0.f;
#pragma unroll
  for (int dy = -1; dy <= 1; ++dy)
#pragma unroll
    for (int dx = -1; dx <= 1; ++dx) {
      int yy = y + dy, xx = x + dx;
      if (yy >= 0 && yy < Hn && xx >= 0 && xx < Wn) s += sl[yy * Wn + xx];
    }
  mask[idx] = s * (1.f / 9.f);
}

// fused instance-norm(vox) + 1x1x1 conv 2->16, writes h16 [b][16][T*HW]
__global__ void k_conv1(const float* vox, const float* vmean, const float* vrstd,
                        const float* w1, float* h16) {
  int idx = blockIdx.x * blockDim.x + threadIdx.x;
  if (idx >= Bn * PLANE) return;
  int tn = idx % PLANE;
  int b = idx / PLANE;
  float v0 = vox[(size_t)(b * 2 + 0) * PLANE + tn];
  float v1 = vox[(size_t)(b * 2 + 1) * PLANE + tn];
  float n0 = (v0 - vmean[b * 2 + 0]) * vrstd[b * 2 + 0];
  float n1 = (v1 - vmean[b * 2 + 1]) * vrstd[b * 2 + 1];
  size_t obase = (size_t)b * 16 * PLANE + tn;
#pragma unroll
  for (int o = 0; o < 16; ++o)
    h16[obase + (size_t)o * PLANE] = w1[o * 2] * n0 + w1[o * 2 + 1] * n1;
}

// attention MLP + per-sample weight synthesis: wdyn[b][o][i][d]
__global__ void k_att(const float* pool, const float* fc1, const float* fc2,
                      const float* dynw, float* wdyn) {
  __shared__ float att[Bn][4];
  int tid = threadIdx.x;
  if (tid < Bn) {
    float z1[8];
#pragma unroll
    for (int j = 0; j < 8; ++j) {
      float s = 0.f;
      for (int i = 0; i < 16; ++i) s += pool[tid * 16 + i] * fc1[j * 16 + i];
      z1[j] = fmaxf(s, 0.f);
    }
    float z2[4]; float mx = -1e30f;
#pragma unroll
    for (int k = 0; k < 4; ++k) {
      float s = 0.f;
      for (int j = 0; j < 8; ++j) s += z1[j] * fc2[k * 8 + j];
      z2[k] = s; mx = fmaxf(mx, s);
    }
    float den = 0.f;
#pragma unroll
    for (int k = 0; k < 4; ++k) { z2[k] = expf(z2[k] - mx); den += z2[k]; }
#pragma unroll
    for (int k = 0; k < 4; ++k) att[tid][k] = z2[k] / den;
  }
  __syncthreads();
  for (int idx = tid; idx < Bn * 768; idx += blockDim.x) {
    int b = idx / 768, r = idx % 768;     // r = o*48 + i*3 + d
    float s = 0.f;
#pragma unroll
    for (int k = 0; k < 4; ++k) s += att[b][k] * dynw[k * 768 + r];
    wdyn[idx] = s;
  }
}

// Dynamic temporal conv as f32 WMMA GEMM: per (b,t), out[16 x HW] = W[16x48] * X[48xHW].
// One wave computes a 16x16 tile: all fragments are loaded up front (one wait
// boundary, loads pipeline), then 12 v_wmma_f32_16x16x4_f32 issue back-to-back.
// Zero-padded temporal taps use zero-filled fragments (0*B == zero padding).
// Fused epilogue: 1x1x1 conv 16->2, sigmoid, * mask -> pre-norm output.
__global__ void __launch_bounds__(256) k_dyn(const float* __restrict__ h16,
                                             const float* __restrict__ wdyn,
                                             const float* __restrict__ c3w,
                                             const float* __restrict__ maskp,
                                             float* __restrict__ out) {
  __shared__ float wsh[768];
  __shared__ float c3[32];
  int tid = threadIdx.x;
  int wave = tid >> 5, lane = tid & 31;
  int tileId = blockIdx.x * 8 + wave;          // < B*T*1024 = 73728
  int b = tileId / (Tn * 1024);
  int rem = tileId - b * (Tn * 1024);
  int t = rem >> 10;
  int n0 = (rem & 1023) << 4;
  for (int i = tid; i < 768; i += 256) wsh[i] = wdyn[b * 768 + i];
  if (tid < 32) c3[tid] = c3w[tid];
  __syncthreads();

  int m = lane & 15;
  int klo = (lane < 16) ? 0 : 2;               // A/B f32 16x16x4 K-layout per ISA
  int t_u = __builtin_amdgcn_readfirstlane(t); // scalar conditions -> EXEC stays full

  // stage 1: gather all A (LDS) and B (global) fragments
  v2f afrag[12], bfrag[12];
#pragma unroll
  for (int c = 0; c < 12; ++c) {
    int d = c >> 2;                            // temporal tap 0..2
    int ib = (c & 3) << 2;                     // input-channel base 0,4,8,12
    int tsrc = t_u + d - 1;
    v2f a = {0.f, 0.f}, bb = {0.f, 0.f};
    if (tsrc >= 0 && tsrc < Tn) {              // scalar branch (t_u is SGPR)
      a.x = wsh[m * 48 + (ib + klo) * 3 + d];
      a.y = wsh[m * 48 + (ib + klo + 1) * 3 + d];
      const float* bp = h16 + ((size_t)(b * 16 + ib) * Tn + tsrc) * HW + n0 + m;
      bb.x = bp[(size_t)klo * PLANE];
      bb.y = bp[(size_t)(klo + 1) * PLANE];
    }
    afrag[c] = a;
    bfrag[c] = bb;
  }

  // stage 2: dense accumulation chain on the matrix pipe
  v8f acc = {};
#pragma unroll
  for (int c = 0; c < 12; ++c)
    acc = __builtin_amdgcn_wmma_f32_16x16x4_f32(false, afrag[c], false, bfrag[c],
                                                (short)0, acc, false, false);

  // D layout: VGPR r -> M=r (lanes 0-15), M=r+8 (lanes 16-31); combine halves.
  int mbase = (lane < 16) ? 0 : 8;
  float p0 = 0.f, p1 = 0.f;
#pragma unroll
  for (int r = 0; r < 8; ++r) {
    float v = acc[r];
    p0 += c3[mbase + r] * v;        // out channel 0
    p1 += c3[16 + mbase + r] * v;   // out channel 1
  }
  p0 += __shfl_xor(p0, 16, 32);
  p1 += __shfl_xor(p1, 16, 32);
  int cch = (lane < 16) ? 0 : 1;
  float s = (lane < 16) ? p0 : p1;
  size_t oidx = ((size_t)(b * 2 + cch) * Tn + t) * HW + n0 + m;
  float g = maskp[oidx] / (1.f + __expf(-s));
  out[oidx] = g;
}

__global__ void k_norm(float* out, const float* omean, const float* orstd) {
  int idx = blockIdx.x * blockDim.x + threadIdx.x;
  if (idx >= VOXSZ) return;
  int bc = idx / PLANE;
  out[idx] = (out[idx] - omean[bc]) * orstd[bc];
}

extern "C" void kernel_launch(void* const* d_in, const int* in_sizes, int n_in,
                              void* d_out, int out_size, void* d_ws, size_t ws_size,
                              hipStream_t stream) {
  (void)n_in; (void)out_size; (void)ws_size;
  const float* t       = (const float*)d_in[0];
  const int*   xi      = (const int*)d_in[1];
  const int*   yi      = (const int*)d_in[2];
  const int*   pi      = (const int*)d_in[3];
  const int*   bi      = (const int*)d_in[4];
  const float* conv1_w = (const float*)d_in[5];
  const float* fc1_w   = (const float*)d_in[6];
  const float* fc2_w   = (const float*)d_in[7];
  const float* dyn_w   = (const float*)d_in[8];
  const float* conv3_w = (const float*)d_in[9];
  float* out = (float*)d_out;
  float* ws  = (float*)d_ws;
  int N = in_sizes[0];

  float* tmax   = ws + OFF_TMAX;
  float* vmean  = ws + OFF_VMEAN;
  float* vrstd  = ws + OFF_VRSTD;
  float* pool   = ws + OFF_POOL;
  float* omean  = ws + OFF_OMEAN;
  float* orstd  = ws + OFF_ORSTD;
  float* wdynp  = ws + OFF_WDYN;
  float* vox    = ws + OFF_VOX;
  float* around = ws + OFF_AROUND;
  float* mask   = ws + OFF_MASK;
  float* h16    = ws + OFF_H16;

  k_zero<<<2048, 256, 0, stream>>>(ws, OFF_VOX + VOXSZ);
  k_tmax<<<(N + 255) / 256, 256, 0, stream>>>(t, bi, tmax, N);
  k_vox<<<(N + 255) / 256, 256, 0, stream>>>(t, xi, yi, pi, bi, tmax, vox, N);
  k_stats<<<Bn * 2, 256, 0, stream>>>(vox, vmean, vrstd, PLANE);
  k_around<<<VOXSZ / 256, 256, 0, stream>>>(vox, around);
  k_mask<<<VOXSZ / 256, 256, 0, stream>>>(around, mask);
  k_conv1<<<(Bn * PLANE) / 256, 256, 0, stream>>>(vox, vmean, vrstd, conv1_w, h16);
  k_mean<<<Bn * 16, 256, 0, stream>>>(h16, pool, PLANE);
  k_att<<<1, 256, 0, stream>>>(pool, fc1_w, fc2_w, dyn_w, wdynp);
  k_dyn<<<(Bn * Tn * 1024) / 8, 256, 0, stream>>>(h16, wdynp, conv3_w, mask, out);
  k_stats<<<Bn * 2, 256, 0, stream>>>(out, omean, orstd, PLANE);
  k_norm<<<VOXSZ / 256, 256, 0, stream>>>(out, omean, orstd);
}